// GCNNet_41961830482015
// MI455X (gfx1250) — compile-verified
//
#include <hip/hip_runtime.h>
#include <hip/hip_bf16.h>

typedef __attribute__((ext_vector_type(2))) float v2f;
typedef __attribute__((ext_vector_type(4))) float v4f;
typedef __attribute__((ext_vector_type(8))) float v8f;

#define CDIV(a,b) (((a)+(b)-1)/(b))

__device__ __forceinline__ void atomAddF(float* p, float v) {
  (void)__hip_atomic_fetch_add(p, v, __ATOMIC_RELAXED, __HIP_MEMORY_SCOPE_AGENT);
}

// ---------------- degree / normalization ----------------
__global__ void k_deg_init(float* deg, int n) {
  int i = blockIdx.x * blockDim.x + threadIdx.x;
  if (i < n) deg[i] = 1.0f;  // self-loop contributes 1 to every node
}
__global__ void k_deg_accum(const int* __restrict__ col, float* deg, int E) {
  int e = blockIdx.x * blockDim.x + threadIdx.x;
  if (e < E) atomAddF(&deg[col[e]], 1.0f);
}
__global__ void k_dinv(float* deg, int n) {
  int i = blockIdx.x * blockDim.x + threadIdx.x;
  if (i < n) deg[i] = rsqrtf(deg[i]);  // deg >= 1 always (self-loops)
}

// ---------------- fp32 WMMA GEMM: Out[M,128] = A[M,128] @ W[128,128] ----------------
// One wave = one 16x16 output tile. Block = 256 thr = 8 waves = all 8 N-tiles
// of a 16-row slab. Uses V_WMMA_F32_16X16X4_F32 (full fp32 precision).
__global__ void k_gemm128_wmma(const float* __restrict__ A,
                               const float* __restrict__ W,
                               float* __restrict__ Out, int M) {
  const int mtile = blockIdx.x;
  const int wave  = threadIdx.x >> 5;   // 0..7 -> N tile
  const int lane  = threadIdx.x & 31;
  const int half  = lane >> 4;          // 0: K={0,1}, 1: K={2,3} within a K-step
  const int l16   = lane & 15;

  const int m0 = mtile * 16;
  int mload = m0 + l16;                  // A row this lane feeds (M index)
  if (mload >= M) mload = M - 1;         // defensive clamp (M=100000 is /16)
  const int n = wave * 16 + l16;         // B/D column this lane feeds

  const float* Arow = A + (size_t)mload * 128;

  v8f acc = {};
#pragma unroll
  for (int kt = 0; kt < 32; ++kt) {
    const int kb = kt * 4 + half * 2;
    // A fragment: lane holds A[m][kb], A[m][kb+1]  (ISA 16x4 f32 layout)
    v2f a = *(const v2f*)(Arow + kb);
    // B fragment: lane holds W[kb][n], W[kb+1][n]
    v2f b;
    b.x = W[(size_t)kb * 128 + n];
    b.y = W[(size_t)(kb + 1) * 128 + n];
    acc = __builtin_amdgcn_wmma_f32_16x16x4_f32(
        /*neg_a=*/false, a, /*neg_b=*/false, b,
        /*c_mod=*/(short)0, acc, /*reuse_a=*/false, /*reuse_b=*/false);
  }

  // D layout: VGPR r, lanes 0-15 -> M=r, lanes 16-31 -> M=r+8; N = lane%16
  const int mbase = m0 + half * 8;
#pragma unroll
  for (int r = 0; r < 8; ++r) {
    const int mr = mbase + r;
    if (mr < M) Out[(size_t)mr * 128 + n] = acc[r];
  }
}

// ---------------- edge scatter: agg[col] += dinv[row]*dinv[col] * h[row] ----------------
// One wave per edge; lane handles 4 contiguous floats of the 128-wide feature.
__global__ void k_scatter(const int* __restrict__ row, const int* __restrict__ col,
                          const float* __restrict__ dinv,
                          const float* __restrict__ h, float* __restrict__ agg,
                          int E, int Etot) {
  const int e = blockIdx.x * (blockDim.x >> 5) + (threadIdx.x >> 5);
  if (e >= Etot) return;
  const int lane = threadIdx.x & 31;
  int r, c;
  if (e < E) { r = row[e]; c = col[e]; }
  else       { r = c = e - E; }          // self-loop edges appended
  const float norm = dinv[r] * dinv[c];
  const float* hs = h   + (size_t)r * 128 + lane * 4;
  float*       as = agg + (size_t)c * 128 + lane * 4;
  v4f v = *(const v4f*)hs;
  atomAddF(as + 0, norm * v.x);
  atomAddF(as + 1, norm * v.y);
  atomAddF(as + 2, norm * v.z);
  atomAddF(as + 3, norm * v.w);
}

// ---------------- bias + relu (in place) ----------------
__global__ void k_bias_relu(float* a, const float* __restrict__ b, size_t total) {
  size_t i = (size_t)blockIdx.x * blockDim.x + threadIdx.x;
  if (i < total) {
    float v = a[i] + b[i & 127];
    a[i] = v > 0.0f ? v : 0.0f;
  }
}

// ---------------- pooling: sums[batch[n]] += h[n], counts[batch[n]] += 1 ----------------
__global__ void k_pool_accum(const float* __restrict__ h, const int* __restrict__ batch,
                             float* sums, float* counts, int n) {
  const int node = blockIdx.x;
  if (node >= n) return;
  const int j = threadIdx.x;                 // 128 threads
  const int g = batch[node];
  atomAddF(&sums[(size_t)g * 128 + j], h[(size_t)node * 128 + j]);
  if (j == 0) atomAddF(&counts[g], 1.0f);
}

// ---------------- per-graph MLP: out = relu(g@Wm1+bm1)@Wm2 + bm2 ----------------
// g = sums/max(counts,1) + b2 (bias of conv2 folded into the mean here).
__global__ void k_mlp(const float* __restrict__ sums, const float* __restrict__ counts,
                      const float* __restrict__ b2,
                      const float* __restrict__ Wm1, const float* __restrict__ bm1,
                      const float* __restrict__ Wm2, const float* __restrict__ bm2,
                      float* __restrict__ out, int G) {
  const int gr = blockIdx.x;
  if (gr >= G) return;
  const int t = threadIdx.x;                 // 64 threads
  __shared__ float gfeat[128];
  __shared__ float part[64];
  float cnt = counts[gr];
  cnt = cnt > 1.0f ? cnt : 1.0f;
  const float inv = 1.0f / cnt;
  for (int j = t; j < 128; j += 64)
    gfeat[j] = sums[(size_t)gr * 128 + j] * inv + b2[j];
  __syncthreads();
  float acc = bm1[t];
  for (int k = 0; k < 128; ++k) acc += gfeat[k] * Wm1[k * 64 + t];
  acc = acc > 0.0f ? acc : 0.0f;
  part[t] = acc * Wm2[t];
  __syncthreads();
  for (int s = 32; s > 0; s >>= 1) {
    if (t < s) part[t] += part[t + s];
    __syncthreads();
  }
  if (t == 0) out[gr] = part[0] + bm2[0];
}

extern "C" void kernel_launch(void* const* d_in, const int* in_sizes, int n_in,
                              void* d_out, int out_size, void* d_ws, size_t ws_size,
                              hipStream_t stream) {
  const float* x    = (const float*)d_in[0];
  const int*   edge = (const int*)d_in[1];
  const int*   batch= (const int*)d_in[2];
  const float* W1   = (const float*)d_in[3];
  const float* b1   = (const float*)d_in[4];
  const float* W2   = (const float*)d_in[5];
  const float* b2   = (const float*)d_in[6];
  const float* Wm1  = (const float*)d_in[7];
  const float* bm1  = (const float*)d_in[8];
  const float* Wm2  = (const float*)d_in[9];
  const float* bm2  = (const float*)d_in[10];

  const int N = in_sizes[0] / 128;     // 100000
  const int E = in_sizes[1] / 2;       // 1600000
  const int G = out_size;              // 512
  const int Etot = E + N;
  const size_t feat = (size_t)N * 128;

  const int* row = edge;
  const int* col = edge + E;

  char* ws = (char*)d_ws;
  size_t off = 0;
  auto alloc = [&](size_t bytes) -> char* {
    char* p = ws + off;
    off += (bytes + 255) & ~(size_t)255;
    return p;
  };
  float* dinv   = (float*)alloc((size_t)N * 4);
  float* bufA   = (float*)alloc(feat * 4);
  float* bufB   = (float*)alloc(feat * 4);
  float* sums   = (float*)alloc((size_t)G * 128 * 4);
  float* counts = (float*)alloc((size_t)G * 4);
  (void)ws_size;

  // normalization coefficients
  k_deg_init <<<CDIV(N, 256), 256, 0, stream>>>(dinv, N);
  k_deg_accum<<<CDIV(E, 256), 256, 0, stream>>>(col, dinv, E);
  k_dinv     <<<CDIV(N, 256), 256, 0, stream>>>(dinv, N);

  // conv1: h1 = x@W1 ; agg ; +b1 ; relu
  k_gemm128_wmma<<<CDIV(N, 16), 256, 0, stream>>>(x, W1, bufA, N);
  hipMemsetAsync(bufB, 0, feat * 4, stream);
  k_scatter  <<<CDIV(Etot, 8), 256, 0, stream>>>(row, col, dinv, bufA, bufB, E, Etot);
  k_bias_relu<<<CDIV(feat, 256), 256, 0, stream>>>(bufB, b1, feat);

  // conv2: h2 = a1@W2 ; agg (b2 folded into pooling)
  k_gemm128_wmma<<<CDIV(N, 16), 256, 0, stream>>>(bufB, W2, bufA, N);
  hipMemsetAsync(bufB, 0, feat * 4, stream);
  k_scatter  <<<CDIV(Etot, 8), 256, 0, stream>>>(row, col, dinv, bufA, bufB, E, Etot);

  // scatter-mean pooling + MLP head
  hipMemsetAsync(sums, 0, (size_t)G * 128 * 4, stream);
  hipMemsetAsync(counts, 0, (size_t)G * 4, stream);
  k_pool_accum<<<N, 128, 0, stream>>>(bufB, batch, sums, counts, N);
  k_mlp<<<G, 64, 0, stream>>>(sums, counts, b2, Wm1, bm1, Wm2, bm2, (float*)d_out, G);
}